// VoxelRasterizer_47536698032167
// MI455X (gfx1250) — compile-verified
//
#include <hip/hip_runtime.h>

#define N_VOXELS  16384
#define N_RAYS    2048
#define THREADS   256
#define CHUNK     256
#define NCHUNKS   (N_VOXELS / CHUNK)
#define MAX_HITS  2048
#define SAMPLES   8
#define STOP_T    0.01f
#define FAR_PLANE 100.0f

#if defined(__HIP_DEVICE_COMPILE__) && __has_builtin(__builtin_amdgcn_global_load_async_to_lds_b128)
#define HAVE_ASYNC_LDS 1
#else
#define HAVE_ASYNC_LDS 0
#endif

typedef int v4i __attribute__((ext_vector_type(4)));
#if HAVE_ASYNC_LDS
typedef __attribute__((address_space(1))) v4i GV4;   // global
typedef __attribute__((address_space(3))) v4i LV4;   // LDS
#endif

__device__ __forceinline__ void wait_async_lds() {
#if HAVE_ASYNC_LDS
    asm volatile("s_wait_asynccnt 0x0" ::: "memory");
#endif
}

// Stage one chunk: 256 voxels of position (3072 B) + size (1024 B) into a
// 4096 B LDS buffer. Exactly one 16 B transfer per thread:
//   tid <  192 -> 16 B of the positions region
//   tid >= 192 -> 16 B of the sizes region
__device__ __forceinline__ void issue_stage(const float* __restrict__ pos,
                                            const float* __restrict__ siz,
                                            int vb, float* dst, int tid) {
    const char* gsrc;
    char* ldst = (char*)dst;
    if (tid < 192) {
        gsrc = (const char*)pos + (size_t)vb * 12 + (size_t)tid * 16;
        ldst += tid * 16;
    } else {
        gsrc = (const char*)siz + (size_t)vb * 4 + (size_t)(tid - 192) * 16;
        ldst += 3072 + (tid - 192) * 16;
    }
#if HAVE_ASYNC_LDS
    __builtin_amdgcn_global_load_async_to_lds_b128((GV4*)gsrc, (LV4*)ldst,
                                                   /*offset=*/0, /*cpol=*/0);
#else
    *(float4*)ldst = *(const float4*)gsrc;
#endif
}

__global__ __launch_bounds__(THREADS)
void VoxelRasterizer_kernel(const float* __restrict__ positions,
                            const float* __restrict__ sizes,
                            const float* __restrict__ densities,
                            const float* __restrict__ colors,
                            const float* __restrict__ ray_o,
                            const float* __restrict__ ray_d,
                            float* __restrict__ out) {
    __shared__ alignas(16) float  s_stage[2][CHUNK * 4];  // [0,768) pos, [768,1024) size
    __shared__ unsigned long long s_key[MAX_HITS];        // (sortable t_near << 32) | slot
    __shared__ float              s_tn[MAX_HITS];
    __shared__ float              s_op[MAX_HITS];
    __shared__ int                s_vid[MAX_HITS];
    __shared__ int                s_cnt;
    __shared__ int                s_n;

    const int ray = blockIdx.x;
    const int tid = threadIdx.x;
    if (tid == 0) s_cnt = 0;

    const float ox = ray_o[ray * 3 + 0], oy = ray_o[ray * 3 + 1], oz = ray_o[ray * 3 + 2];
    const float dx = ray_d[ray * 3 + 0], dy = ray_d[ray * 3 + 1], dz = ray_d[ray * 3 + 2];
    const float ix = 1.0f / dx, iy = 1.0f / dy, iz = 1.0f / dz;

    issue_stage(positions, sizes, 0, &s_stage[0][0], tid);
    __syncthreads();  // covers s_cnt init

    for (int c = 0; c < NCHUNKS; ++c) {
        wait_async_lds();
        __syncthreads();  // stage[c&1] fully resident; prev buffer fully consumed
        if (c + 1 < NCHUNKS)
            issue_stage(positions, sizes, (c + 1) * CHUNK, &s_stage[(c + 1) & 1][0], tid);

        const float* buf = &s_stage[c & 1][0];
        const int v = c * CHUNK + tid;
        const float px = buf[3 * tid + 0];
        const float py = buf[3 * tid + 1];
        const float pz = buf[3 * tid + 2];
        const float h  = 0.5f * buf[3 * CHUNK + tid];

        const float t0x = (px - h - ox) * ix, t1x = (px + h - ox) * ix;
        const float t0y = (py - h - oy) * iy, t1y = (py + h - oy) * iy;
        const float t0z = (pz - h - oz) * iz, t1z = (pz + h - oz) * iz;
        const float tn = fmaxf(fmaxf(fminf(t0x, t1x), fminf(t0y, t1y)), fminf(t0z, t1z));
        const float tf = fminf(fminf(fmaxf(t0x, t1x), fmaxf(t0y, t1y)), fmaxf(t0z, t1z));

        if (tf > tn && tf > 0.0f) {
            const int slot = atomicAdd(&s_cnt, 1);
            if (slot < MAX_HITS) {
                const float delta = (tf - tn) * (1.0f / (float)(SAMPLES - 1));
                const float opv   = 1.0f - __expf(-__expf(densities[v]) * delta);
                s_tn[slot]  = tn;
                s_op[slot]  = opv;
                s_vid[slot] = v;
                const unsigned ub = __float_as_uint(tn);
                const unsigned sk = ub ^ (unsigned)(((int)ub >> 31) | 0x80000000);
                s_key[slot] = ((unsigned long long)sk << 32) | (unsigned)slot;
            }
        }
    }
    __syncthreads();

    if (tid == 0) s_n = (s_cnt < MAX_HITS) ? s_cnt : MAX_HITS;
    __syncthreads();
    const int n = s_n;

    if (n > 0) {
        int P = 1;
        while (P < n) P <<= 1;
        for (int i = n + tid; i < P; i += THREADS) s_key[i] = ~0ULL;
        __syncthreads();
        // Bitonic sort of P 64-bit keys (ascending t_near).
        for (int k = 2; k <= P; k <<= 1) {
            for (int j = k >> 1; j > 0; j >>= 1) {
                for (int i = tid; i < P; i += THREADS) {
                    const int pj = i ^ j;
                    if (pj > i) {
                        const unsigned long long a = s_key[i];
                        const unsigned long long b = s_key[pj];
                        const bool up = ((i & k) == 0);
                        if ((a > b) == up) { s_key[i] = b; s_key[pj] = a; }
                    }
                }
                __syncthreads();
            }
        }
    }

    if (tid == 0) {
        float r = 0.f, g = 0.f, b = 0.f, d = 0.f, ws = 0.f, T = 1.0f;
        for (int i = 0; i < n; ++i) {
            if (T < STOP_T) break;  // exact: T_excl monotone non-increasing
            const unsigned long long key = s_key[i];
            const int   slot = (int)(unsigned)(key & 0xFFFFFFFFu);
            const float opv  = s_op[slot];
            const float w    = T * opv;
            const int   v    = s_vid[slot];
            r  += w * colors[3 * v + 0];
            g  += w * colors[3 * v + 1];
            b  += w * colors[3 * v + 2];
            d  += w * s_tn[slot];
            ws += w;
            T  *= (1.0f - opv);
        }
        out[ray * 3 + 0]      = r;
        out[ray * 3 + 1]      = g;
        out[ray * 3 + 2]      = b;
        out[N_RAYS * 3 + ray] = (n > 0) ? d : FAR_PLANE;  // depth
        out[N_RAYS * 4 + ray] = ws;                       // weights
        // T_final * BG_COLOR omitted: BG_COLOR == 0.
    }
}

extern "C" void kernel_launch(void* const* d_in, const int* in_sizes, int n_in,
                              void* d_out, int out_size, void* d_ws, size_t ws_size,
                              hipStream_t stream) {
    (void)in_sizes; (void)n_in; (void)out_size; (void)d_ws; (void)ws_size;
    const float* positions = (const float*)d_in[0];
    const float* sizes     = (const float*)d_in[1];
    const float* densities = (const float*)d_in[2];
    const float* colors    = (const float*)d_in[3];
    const float* ray_o     = (const float*)d_in[4];
    const float* ray_d     = (const float*)d_in[5];
    VoxelRasterizer_kernel<<<N_RAYS, THREADS, 0, stream>>>(
        positions, sizes, densities, colors, ray_o, ray_d, (float*)d_out);
}